// sar_model_39857296507426
// MI455X (gfx1250) — compile-verified
//
#include <hip/hip_runtime.h>

typedef __attribute__((ext_vector_type(2))) float v2f;
typedef __attribute__((ext_vector_type(4))) float v4f;
typedef __attribute__((ext_vector_type(8))) float v8f;

#define TWO_PI 6.28318530717958647692f

// ---------------------------------------------------------------------------
// Build combined (fftshift o DFT o fftshift) matrix, optionally transposed in
// storage, zero-padded to ld x ld. sign=-1 forward, +1 inverse (scale=1/N).
// Exact integer phase: angle = sign * 2*pi * ((k2*mm) mod N) / N.
// ---------------------------------------------------------------------------
__global__ void build_dft_kernel(float* __restrict__ Wre, float* __restrict__ Wim,
                                 int N, int ld, int sign, float scale, int transpose)
{
    int i = blockIdx.x * blockDim.x + threadIdx.x;
    if (i >= ld * ld) return;
    int r = i / ld, c = i % ld;
    float vr = 0.f, vi = 0.f;
    if (r < N && c < N) {
        int k = transpose ? c : r;   // output (frequency) index of logical matrix
        int m = transpose ? r : c;   // input (sample) index
        int h = N >> 1;
        int k2 = (k + h) % N;
        int mm = (m - h) % N; if (mm < 0) mm += N;
        int e = (int)(((long long)k2 * (long long)mm) % (long long)N);
        float ang = (float)sign * TWO_PI * (float)e / (float)N;
        vr = scale * cosf(ang);
        vi = scale * sinf(ang);
    }
    Wre[i] = vr;
    Wim[i] = vi;
}

// ---------------------------------------------------------------------------
// Phase compensation: U = (fr + i*fi) * (cos(ph) - i*sin(ph)),
// ph = 2*pi * f_seg[f] * (2*Z0/C), Z0=2, C=3e8.
// ---------------------------------------------------------------------------
__global__ void compensate_kernel(const float* __restrict__ fr,
                                  const float* __restrict__ fi,
                                  const float* __restrict__ fseg,
                                  float* __restrict__ Ure, float* __restrict__ Uim,
                                  int total, int NF)
{
    int i = blockIdx.x * blockDim.x + threadIdx.x;
    if (i >= total) return;
    int f = i % NF;
    float ph = TWO_PI * fseg[f] * (4.0f / 3.0e8f);   // 2*Z0/C, Z0 = 2
    float c = cosf(ph), s = sinf(ph);
    float a = fr[i], b = fi[i];
    Ure[i] = a * c + b * s;
    Uim[i] = b * c - a * s;
}

// ---------------------------------------------------------------------------
// Stolt interpolation: per pixel p, resample S[p,:] from nonuniform kz[p,:]
// onto the uniform kz_interp grid (clamped at the ends like jnp.interp).
// ---------------------------------------------------------------------------
__global__ void stolt_kernel(const float* __restrict__ Sre, const float* __restrict__ Sim,
                             const float* __restrict__ kz, const float* __restrict__ kzi,
                             float* __restrict__ Ore, float* __restrict__ Oim,
                             int P, int NF)
{
    int i = blockIdx.x * blockDim.x + threadIdx.x;
    if (i >= P * NF) return;
    int p = i / NF, j = i % NF;
    long long base = (long long)p * NF;
    const float* xp = kz + base;
    float t = kzi[j];
    float re, im;
    if (t <= xp[0]) {
        re = Sre[base]; im = Sim[base];
    } else if (t >= xp[NF - 1]) {
        re = Sre[base + NF - 1]; im = Sim[base + NF - 1];
    } else {
        int lo = 0, hi = NF - 1;
        while (hi - lo > 1) {
            int mid = (lo + hi) >> 1;
            if (xp[mid] <= t) lo = mid; else hi = mid;
        }
        float x0 = xp[lo], x1 = xp[lo + 1];
        float w = (x1 > x0) ? (t - x0) / (x1 - x0) : 0.f;
        float r0 = Sre[base + lo], r1 = Sre[base + lo + 1];
        float i0 = Sim[base + lo], i1 = Sim[base + lo + 1];
        re = r0 + w * (r1 - r0);
        im = i0 + w * (i1 - i0);
    }
    Ore[i] = re;
    Oim[i] = im;
}

// ---------------------------------------------------------------------------
// Complex GEMM C = A @ B on V_WMMA_F32_16X16X4_F32.
// Planar complex operands (separate re/im planes). 128 threads = 4 wave32 in
// a 2x2 wave grid; each wave owns a 32x32 register tile (2x2 fragments, 4
// complex accumulators) -> 64x64 block tile. K staged in 32-deep LDS chunks,
// global->LDS staging uses b128 when the leading dimension permits.
// C element stride cstr allows interleaved complex output.
// Batched via gridDim.z with per-operand batch strides (in elements).
// ---------------------------------------------------------------------------
#define KC 32
#define APAD 36   // 36*4B = 144B row pitch: 16B aligned, conflict-free frag reads
#define BPAD 68   // 68*4B = 272B row pitch: 16B aligned

__global__ __launch_bounds__(128)
void cgemm_wmma(const float* __restrict__ Ar, const float* __restrict__ Ai,
                long long sAb, int lda,
                const float* __restrict__ Br, const float* __restrict__ Bi,
                long long sBb, int ldb,
                float* __restrict__ Cr, float* __restrict__ Ci,
                long long sCb, int ldc, int cstr,
                int M, int N, int K)
{
    __shared__ float As[2][64][APAD];  // [re/im][m][k]
    __shared__ float Bs[2][KC][BPAD];  // [re/im][k][n]

    const int tid  = threadIdx.x;
    const int wid  = tid >> 5;
    const int lane = tid & 31;
    const int bm = blockIdx.y * 64;
    const int bn = blockIdx.x * 64;

    const float* Arb = Ar + (long long)blockIdx.z * sAb;
    const float* Aib = Ai + (long long)blockIdx.z * sAb;
    const float* Brb = Br + (long long)blockIdx.z * sBb;
    const float* Bib = Bi + (long long)blockIdx.z * sBb;

    const bool vecA = ((lda & 3) == 0);
    const bool vecB = ((ldb & 3) == 0);

    v8f cre[2][2] = {};
    v8f cim[2][2] = {};

    const int mbase = ((wid >> 1) * 32) + (lane & 15);  // A-frag row base in block tile
    const int nbase = ((wid & 1) * 32) + (lane & 15);   // B-frag col base in block tile
    const int ksel  = (lane >> 4) * 2;                  // K sub-slot per ISA layout

    for (int k0 = 0; k0 < K; k0 += KC) {
        // ---- Cooperative chunk load (zero-filled out of range) ----
        // A chunk: 64 x KC per plane, 4 consecutive k per thread.
        for (int q = tid * 4; q < 64 * KC; q += 128 * 4) {
            int mi = q >> 5;          // KC == 32
            int ki = q & 31;
            int gm = bm + mi, gk = k0 + ki;
            long long ia = (long long)gm * lda + gk;
            if (vecA && gm < M && (gk + 3) < K) {
                *(v4f*)&As[0][mi][ki] = *(const v4f*)&Arb[ia];
                *(v4f*)&As[1][mi][ki] = *(const v4f*)&Aib[ia];
            } else {
#pragma unroll
                for (int u = 0; u < 4; ++u) {
                    bool ok = (gm < M) && ((gk + u) < K);
                    As[0][mi][ki + u] = ok ? Arb[ia + u] : 0.f;
                    As[1][mi][ki + u] = ok ? Aib[ia + u] : 0.f;
                }
            }
        }
        // B chunk: KC x 64 per plane, 4 consecutive n per thread.
        for (int q = tid * 4; q < KC * 64; q += 128 * 4) {
            int ri = q >> 6;
            int ci = q & 63;
            int gk = k0 + ri, gn = bn + ci;
            long long ib = (long long)gk * ldb + gn;
            if (vecB && gk < K && (gn + 3) < N) {
                *(v4f*)&Bs[0][ri][ci] = *(const v4f*)&Brb[ib];
                *(v4f*)&Bs[1][ri][ci] = *(const v4f*)&Bib[ib];
            } else {
#pragma unroll
                for (int u = 0; u < 4; ++u) {
                    bool ok = (gk < K) && ((gn + u) < N);
                    Bs[0][ri][ci + u] = ok ? Brb[ib + u] : 0.f;
                    Bs[1][ri][ci + u] = ok ? Bib[ib + u] : 0.f;
                }
            }
        }
        __syncthreads();

        // Speculative prefetch of next K chunk (global_prefetch_b8).
        if (k0 + KC < K) {
            __builtin_prefetch(Arb + (long long)(bm + mbase) * lda + (k0 + KC), 0, 1);
            __builtin_prefetch(Brb + (long long)(k0 + KC) * ldb + bn + lane, 0, 1);
        }

#pragma unroll
        for (int kk = 0; kk < KC; kk += 4) {
            int ka = kk + ksel;
            v2f ar[2], ai[2], ain[2], br[2], bi[2];
#pragma unroll
            for (int t = 0; t < 2; ++t) {
                int mr = mbase + t * 16;
                ar[t].x = As[0][mr][ka];  ar[t].y = As[0][mr][ka + 1];
                ai[t].x = As[1][mr][ka];  ai[t].y = As[1][mr][ka + 1];
                ain[t] = -ai[t];          // f32 WMMA has no A-neg modifier
                int nc = nbase + t * 16;
                br[t].x = Bs[0][ka][nc];  br[t].y = Bs[0][ka + 1][nc];
                bi[t].x = Bs[1][ka][nc];  bi[t].y = Bs[1][ka + 1][nc];
            }
#pragma unroll
            for (int mi = 0; mi < 2; ++mi) {
#pragma unroll
                for (int ni = 0; ni < 2; ++ni) {
                    // Cre += Ar*Br + (-Ai)*Bi ; Cim += Ar*Bi + Ai*Br
                    cre[mi][ni] = __builtin_amdgcn_wmma_f32_16x16x4_f32(false, ar[mi],  false, br[ni], (short)0, cre[mi][ni], false, false);
                    cre[mi][ni] = __builtin_amdgcn_wmma_f32_16x16x4_f32(false, ain[mi], false, bi[ni], (short)0, cre[mi][ni], false, false);
                    cim[mi][ni] = __builtin_amdgcn_wmma_f32_16x16x4_f32(false, ar[mi],  false, bi[ni], (short)0, cim[mi][ni], false, false);
                    cim[mi][ni] = __builtin_amdgcn_wmma_f32_16x16x4_f32(false, ai[mi],  false, br[ni], (short)0, cim[mi][ni], false, false);
                }
            }
        }
        __syncthreads();
    }

    // Store D per ISA C/D layout: vgpr v, lanes<16 -> M=v, lanes>=16 -> M=v+8.
    long long cb = (long long)blockIdx.z * sCb;
#pragma unroll
    for (int mi = 0; mi < 2; ++mi) {
#pragma unroll
        for (int ni = 0; ni < 2; ++ni) {
            int crow = bm + ((wid >> 1) * 32) + mi * 16 + ((lane >> 4) * 8);
            int ccol = bn + ((wid & 1) * 32) + ni * 16 + (lane & 15);
            if (ccol < N) {
#pragma unroll
                for (int v = 0; v < 8; ++v) {
                    int r = crow + v;
                    if (r < M) {
                        long long idx = cb + (long long)r * ldc + ccol;
                        Cr[idx * cstr] = cre[mi][ni][v];
                        Ci[idx * cstr] = cim[mi][ni][v];
                    }
                }
            }
        }
    }
}

// ---------------------------------------------------------------------------
// Host-side orchestration.
// ---------------------------------------------------------------------------
extern "C" void kernel_launch(void* const* d_in, const int* in_sizes, int n_in,
                              void* d_out, int out_size, void* d_ws, size_t ws_size,
                              hipStream_t stream)
{
    (void)n_in; (void)out_size; (void)ws_size;

    const float* fs_re = (const float*)d_in[0];
    const float* fs_im = (const float*)d_in[1];
    const float* kz    = (const float*)d_in[2];
    const float* kzi   = (const float*)d_in[3];
    const float* fseg  = (const float*)d_in[4];

    const int NF = in_sizes[3];          // 375
    const int NX = 200, NY = 200;        // fixed aperture geometry
    const int P  = NX * NY;              // 40000 pixels
    const long long TOT = (long long)P * NF;
    const int LD200 = 208;               // 200 padded to 16
    const int LDZ   = ((NF + 15) / 16) * 16;   // 384

    float* ws = (float*)d_ws;
    size_t o = 0;
    float* Wf_re = ws + o; o += (size_t)LD200 * LD200;
    float* Wf_im = ws + o; o += (size_t)LD200 * LD200;
    float* Wi_re = ws + o; o += (size_t)LD200 * LD200;
    float* Wi_im = ws + o; o += (size_t)LD200 * LD200;
    float* Wz_re = ws + o; o += (size_t)LDZ * LDZ;
    float* Wz_im = ws + o; o += (size_t)LDZ * LDZ;
    float* Ure = ws + o; o += (size_t)TOT;
    float* Uim = ws + o; o += (size_t)TOT;
    float* Vre = ws + o; o += (size_t)TOT;
    float* Vim = ws + o; o += (size_t)TOT;

    // 1) DFT matrices (forward x/y, inverse x/y, inverse z stored transposed).
    {
        int n = LD200 * LD200;
        build_dft_kernel<<<(n + 255) / 256, 256, 0, stream>>>(Wf_re, Wf_im, NX, LD200, -1, 1.0f, 0);
        build_dft_kernel<<<(n + 255) / 256, 256, 0, stream>>>(Wi_re, Wi_im, NX, LD200, +1, 1.0f / NX, 0);
        int nz = LDZ * LDZ;
        build_dft_kernel<<<(nz + 255) / 256, 256, 0, stream>>>(Wz_re, Wz_im, NF, LDZ, +1, 1.0f / NF, 1);
    }

    // 2) Phase compensation into U.
    compensate_kernel<<<((int)TOT + 255) / 256, 256, 0, stream>>>(
        fs_re, fs_im, fseg, Ure, Uim, (int)TOT, NF);

    dim3 blk(128, 1, 1);

    // 3) Forward FFT along x: per-y batch, V_y = Wf @ U_y  (200x375 = 200x200 @ 200x375)
    {
        dim3 grid((NF + 63) / 64, (NX + 63) / 64, NY);
        cgemm_wmma<<<grid, blk, 0, stream>>>(Wf_re, Wf_im, 0, LD200,
                                             Ure, Uim, (long long)NX * NF, NF,
                                             Vre, Vim, (long long)NX * NF, NF, 1,
                                             NX, NF, NX);
    }
    // 4) Forward FFT along y: U = Wf @ V viewed as (NY x NX*NF)
    {
        int NN = NX * NF;
        dim3 grid((NN + 63) / 64, (NY + 63) / 64, 1);
        cgemm_wmma<<<grid, blk, 0, stream>>>(Wf_re, Wf_im, 0, LD200,
                                             Vre, Vim, 0, NN,
                                             Ure, Uim, 0, NN, 1,
                                             NY, NN, NY);
    }
    // 5) Stolt interpolation along frequency: U -> V
    stolt_kernel<<<((int)TOT + 255) / 256, 256, 0, stream>>>(
        Ure, Uim, kz, kzi, Vre, Vim, P, NF);

    // 6) Inverse FFT along y: U = Wi @ V
    {
        int NN = NX * NF;
        dim3 grid((NN + 63) / 64, (NY + 63) / 64, 1);
        cgemm_wmma<<<grid, blk, 0, stream>>>(Wi_re, Wi_im, 0, LD200,
                                             Vre, Vim, 0, NN,
                                             Ure, Uim, 0, NN, 1,
                                             NY, NN, NY);
    }
    // 7) Inverse FFT along x: per-y batch, V_y = Wi @ U_y
    {
        dim3 grid((NF + 63) / 64, (NX + 63) / 64, NY);
        cgemm_wmma<<<grid, blk, 0, stream>>>(Wi_re, Wi_im, 0, LD200,
                                             Ure, Uim, (long long)NX * NF, NF,
                                             Vre, Vim, (long long)NX * NF, NF, 1,
                                             NX, NF, NX);
    }
    // 8) Inverse FFT along z: out = V (P x NF) @ Wz^T, interleaved complex64 out.
    {
        float* out = (float*)d_out;
        dim3 grid((NF + 63) / 64, (P + 63) / 64, 1);
        cgemm_wmma<<<grid, blk, 0, stream>>>(Vre, Vim, 0, NF,
                                             Wz_re, Wz_im, 0, LDZ,
                                             out, out + 1, 0, NF, 2,
                                             P, NF, NF);
    }
}